// Layer1_VertexUpdate_91096256348923
// MI455X (gfx1250) — compile-verified
//
#include <hip/hip_runtime.h>

// -----------------------------------------------------------------------------
// segment_sum(edge_attr, edgeij_pair[0]) over 1M vertices, then
// out[i] = [A_ii, C_i, gamma_i / A_ii]   (row-major (N,3) f32)
//
// MI455X reasoning: bandwidth-bound scatter-add.
//   - 128 MB edge stream (idx32 + f32) read once  -> ~5.5 us @ 23.3 TB/s floor
//   - 4 MB accumulator lives in 192 MB L2; 16M non-returning
//     global_atomic_add_f32 (inline asm, scope:SCOPE_DEV) resolve in L2's
//     atomic units (STOREcnt path) and never round-trip to HBM.
//   - TH_LOAD_NT 128-bit loads keep the one-shot stream from evicting the
//     accumulator; global_prefetch_b8 hides stream latency.
//   - finalize stages 3-wide rows in LDS, then uses the CDNA5 async engine
//     (global_store_async_from_lds_b128, ASYNCcnt) to move LDS->global
//     directly, skipping the LDS->VGPR->memory round trip.
// -----------------------------------------------------------------------------

typedef __attribute__((ext_vector_type(4))) int   i32x4;
typedef __attribute__((ext_vector_type(4))) float f32x4;
typedef __attribute__((ext_vector_type(2))) float f32x2;

#define THREADS 256
#define EDGES_PER_THREAD 4
// Prefetch distance in elements (~64 KB ahead on each stream)
#define PF_DIST 16384

// Guaranteed single hardware FP32 atomic, non-returning form (TH[0]=0 ->
// tracked by STOREcnt), device scope so the add is performed in L2.
__device__ __forceinline__ void atomic_add_f32_noret(float* p, float v) {
  asm volatile("global_atomic_add_f32 %0, %1, off scope:SCOPE_DEV"
               :: "v"(p), "v"(v)
               : "memory");
}

// ---------------------------------------------------------------------------
// Kernel 1: streaming scatter-add, 4 edges per thread via 128-bit NT loads.
// ---------------------------------------------------------------------------
__global__ __launch_bounds__(THREADS) void scatter_add4(
    const int* __restrict__ src_idx,   // row 0 of edgeij_pair (int32 per harness)
    const float* __restrict__ eattr,   // edge weights
    float* __restrict__ gamma,         // accumulator (d_ws), pre-zeroed
    int n_edges) {
  const int t = blockIdx.x * blockDim.x + threadIdx.x;
  const int base = t * EDGES_PER_THREAD;

  if (base + EDGES_PER_THREAD <= n_edges) {
    // Speculative prefetch of the stream well ahead (global_prefetch_b8).
    if (base + PF_DIST < n_edges) {
      __builtin_prefetch(src_idx + base + PF_DIST, 0, 0);
      __builtin_prefetch(eattr + base + PF_DIST, 0, 0);
    }
    // Non-temporal 128-bit loads: stream data, do not pollute L2 (which holds
    // the hot 4 MB accumulator).
    i32x4 idx = __builtin_nontemporal_load((const i32x4*)(src_idx + base));
    f32x4 v   = __builtin_nontemporal_load((const f32x4*)(eattr + base));

    atomic_add_f32_noret(&gamma[idx.x], v.x);
    atomic_add_f32_noret(&gamma[idx.y], v.y);
    atomic_add_f32_noret(&gamma[idx.z], v.z);
    atomic_add_f32_noret(&gamma[idx.w], v.w);
  } else {
    // Tail (not hit for 16M edges, but keep it correct/generic).
    for (int e = base; e < n_edges; ++e) {
      atomic_add_f32_noret(&gamma[src_idx[e]], eattr[e]);
    }
  }
}

// ---------------------------------------------------------------------------
// Kernel 2: finalize. Each thread handles one vertex; rows staged in LDS,
// then the async engine DMAs the tile LDS->global as 128-bit transfers.
// ---------------------------------------------------------------------------
__global__ __launch_bounds__(THREADS) void finalize3(
    const f32x2* __restrict__ vattr,   // (N,2) interleaved [A_ii, C_i]
    const float* __restrict__ gamma,
    float* __restrict__ out,           // (N,3) row-major
    int n) {
  __shared__ float tile[3 * THREADS];

  const int blockBase = blockIdx.x * THREADS;
  const int i = blockBase + threadIdx.x;

  float a = 1.0f, c = 0.0f, g = 0.0f;
  if (i < n) {
    f32x2 ac = vattr[i];            // coalesced b64 load
    a = ac.x;
    c = ac.y;
    g = gamma[i];
  }
  tile[3 * threadIdx.x + 0] = a;
  tile[3 * threadIdx.x + 1] = c;
  tile[3 * threadIdx.x + 2] = g / a;  // alpha_i = gammabar / A_ii
  __syncthreads();   // s_wait_dscnt + barrier: tile fully written

  // 3*THREADS floats per block; blockBase*3*4 bytes is always 16B-aligned
  // (3072 B per block). DMA the tile out with the async LDS->global path.
  const int outBase  = blockBase * 3;
  const int totalOut = 3 * n;
  const int nQuads   = (3 * THREADS) / 4;     // 192
  for (int j = threadIdx.x; j < nQuads; j += THREADS) {
    const int fo = outBase + 4 * j;
    if (fo + 3 < totalOut) {
      // LDS flat addresses carry the LDS byte offset in their low 32 bits.
      unsigned ldsOff = (unsigned)(unsigned long long)(tile + 4 * j);
      asm volatile("global_store_async_from_lds_b128 %0, %1, off"
                   :: "v"(out + fo), "v"(ldsOff)
                   : "memory");
    } else {
      for (int k = 0; k < 4; ++k)
        if (fo + k < totalOut) out[fo + k] = tile[4 * j + k];
    }
  }
  // Drain async LDS->global transfers before wave termination.
  asm volatile("s_wait_asynccnt 0x0" ::: "memory");
}

// ---------------------------------------------------------------------------
// Host-side launcher (graph-capture safe: only async ops on `stream`).
// ---------------------------------------------------------------------------
extern "C" void kernel_launch(void* const* d_in, const int* in_sizes, int n_in,
                              void* d_out, int out_size, void* d_ws, size_t ws_size,
                              hipStream_t stream) {
  (void)n_in; (void)out_size; (void)ws_size;

  const float* vattr = (const float*)d_in[0];        // (N,2) f32
  const int*   eij   = (const int*)d_in[1];          // (2,E) int32; row 0 first
  const float* eattr = (const float*)d_in[2];        // (E,1) f32
  // d_in[3] (g) and d_in[4] (batch) are unused by the reference math.

  const int n_vertices = in_sizes[0] / 2;
  const int n_edges    = in_sizes[2];

  float* gamma = (float*)d_ws;                        // n_vertices floats
  float* out   = (float*)d_out;                       // 3*n_vertices floats

  // Zero the accumulator every call (scatter kernel accumulates into it).
  hipMemsetAsync(gamma, 0, (size_t)n_vertices * sizeof(float), stream);

  const int nThreadsScatter = (n_edges + EDGES_PER_THREAD - 1) / EDGES_PER_THREAD;
  const int scatterBlocks   = (nThreadsScatter + THREADS - 1) / THREADS;
  scatter_add4<<<scatterBlocks, THREADS, 0, stream>>>(eij, eattr, gamma, n_edges);

  const int finBlocks = (n_vertices + THREADS - 1) / THREADS;
  finalize3<<<finBlocks, THREADS, 0, stream>>>((const f32x2*)vattr, gamma, out, n_vertices);
}